// GCN_5488968204990
// MI455X (gfx1250) — compile-verified
//
#include <hip/hip_runtime.h>

#define NN 100000      // nodes
#define NE 1600000     // edges
#define NG 5000        // graphs
#define FIN 11
#define HID 128
#define NCLS 19

typedef __attribute__((ext_vector_type(2))) float v2f;
typedef __attribute__((ext_vector_type(8))) float v8f;

__device__ __forceinline__ void atomAddF(float* p, float v) {
    // native f32 atomic add, relaxed, device scope -> global_atomic_add_f32
    __hip_atomic_fetch_add(p, v, __ATOMIC_RELAXED, __HIP_MEMORY_SCOPE_AGENT);
}

// ---------------------------------------------------------------- utility ---
__global__ void k_fill(float* __restrict__ p, float v, int n) {
    int i = blockIdx.x * blockDim.x + threadIdx.x;
    if (i < n) p[i] = v;
}

// pad x [NN x FIN] -> xp [NN x 16] (zero-filled cols 11..15)
__global__ void k_pad_x(const float* __restrict__ x, float* __restrict__ xp) {
    int i = blockIdx.x * blockDim.x + threadIdx.x;
    if (i < NN * 16) {
        int n = i >> 4, k = i & 15;
        xp[i] = (k < FIN) ? x[n * FIN + k] : 0.0f;
    }
}

// deg[dst] += 1 per edge (deg pre-filled with 1.0 for the self loop)
__global__ void k_edge_deg(const long long* __restrict__ dst, float* __restrict__ deg, int E) {
    int e = blockIdx.x * blockDim.x + threadIdx.x;
    if (e < E) atomAddF(&deg[(int)dst[e]], 1.0f);
}

__global__ void k_node_norm(const float* __restrict__ deg, float* __restrict__ dis,
                            float* __restrict__ selfw, int n) {
    int i = blockIdx.x * blockDim.x + threadIdx.x;
    if (i < n) {
        float d  = deg[i];
        dis[i]   = rsqrtf(d);
        selfw[i] = 1.0f / d;   // dis*dis
    }
}

__global__ void k_edge_coef(const long long* __restrict__ src, const long long* __restrict__ dst,
                            const float* __restrict__ dis, float* __restrict__ nrm, int E) {
    int e = blockIdx.x * blockDim.x + threadIdx.x;
    if (e < E) nrm[e] = dis[(int)src[e]] * dis[(int)dst[e]];
}

// ------------------------------------------------------------- WMMA GEMM ----
// H[M x 128] = A[M x K] @ W[KROWS x 128] (rows KROWS..K-1 implicitly zero).
// fp32 WMMA 16x16x4; block = 256 threads = 8 waves; wave -> 16 rows x 128 cols.
// LDS W layout swizzled so each lane's (k, k+1) pair is one ds_load_b64:
//   element (4*kt + 2*h + j, col) at sW[ ((kt*2+h)*128 + col)*2 + j ]
template<int K, int KROWS>
__global__ void __launch_bounds__(256)
k_gemm_wmma(const float* __restrict__ A, const float* __restrict__ W,
            float* __restrict__ H, int M) {
    __shared__ float sW[K * HID];
    const int tid  = threadIdx.x;
    const int lane = tid & 31;
    const int wave = tid >> 5;
    const int l16  = lane & 15;
    const int half = lane >> 4;

    for (int i = tid; i < K * HID; i += 256) {
        const int k = i >> 7, c = i & 127;
        const float v = (KROWS == K || k < KROWS) ? W[k * HID + c] : 0.0f;
        const int kt = k >> 2, h = (k >> 1) & 1, j = k & 1;
        sW[((((kt << 1) + h) << 7) + c) * 2 + j] = v;
    }
    __syncthreads();

    const int row0 = blockIdx.x * 128 + wave * 16;
    if (row0 >= M) return;               // wave-uniform: EXEC stays all-ones

    v8f c[8];
#pragma unroll
    for (int t = 0; t < 8; ++t) c[t] = (v8f){0.f,0.f,0.f,0.f,0.f,0.f,0.f,0.f};

    // A-frag: lane l16 -> row row0+l16; half selects K offsets {0,1} / {2,3}
    const v2f* arow = (const v2f*)(A + (size_t)(row0 + l16) * K);
    const v2f* sW2  = (const v2f*)sW;

#pragma unroll 4
    for (int kt = 0; kt < (K >> 2); ++kt) {
        const v2f a = arow[2 * kt + half];           // one global_load_b64
        const int bbase = ((kt << 1) + half) << 7;
#pragma unroll
        for (int t = 0; t < 8; ++t) {
            const v2f b = sW2[bbase + t * 16 + l16]; // one ds_load_b64
            c[t] = __builtin_amdgcn_wmma_f32_16x16x4_f32(
                false, a, false, b, (short)0, c[t], false, false);
        }
    }

    // D layout: VGPR r, lanes 0-15 -> (M=r, N=lane); lanes 16-31 -> (M=8+r, N=lane-16)
    float* hrow = H + (size_t)row0 * HID;
#pragma unroll
    for (int t = 0; t < 8; ++t) {
        const int col = t * 16 + l16;
#pragma unroll
        for (int r = 0; r < 8; ++r) {
            hrow[(size_t)(half * 8 + r) * HID + col] = c[t][r];
        }
    }
}

// agg = h * self_w[row] + bias[col]   (covers every element: no zero pass needed)
__global__ void k_combine_init(const float* __restrict__ h, const float* __restrict__ selfw,
                               const float* __restrict__ bias, float* __restrict__ agg) {
    size_t i = (size_t)blockIdx.x * blockDim.x + threadIdx.x;
    if (i < (size_t)NN * HID) {
        int row = (int)(i >> 7);
        int col = (int)(i & 127);
        agg[i] = h[i] * selfw[row] + bias[col];
    }
}

// one wave per edge: lane loads float4 of h[src], scales, 4x f32 atomic into agg[dst]
__global__ void __launch_bounds__(256)
k_edge_scatter(const float* __restrict__ h, float* __restrict__ agg,
               const long long* __restrict__ src, const long long* __restrict__ dst,
               const float* __restrict__ nrm, int E) {
    int e = blockIdx.x * 8 + (threadIdx.x >> 5);
    if (e >= E) return;
    const int lane = threadIdx.x & 31;
    const int s = (int)src[e];
    const int d = (int)dst[e];
    const float w = nrm[e];
    const float4 v = ((const float4*)(h + (size_t)s * HID))[lane];
    float* ad = agg + (size_t)d * HID + lane * 4;
    atomAddF(ad + 0, v.x * w);
    atomAddF(ad + 1, v.y * w);
    atomAddF(ad + 2, v.z * w);
    atomAddF(ad + 3, v.w * w);
}

__global__ void k_relu(float* __restrict__ p) {
    size_t i = (size_t)blockIdx.x * blockDim.x + threadIdx.x;
    if (i < (size_t)NN * HID) p[i] = fmaxf(p[i], 0.0f);
}

// one wave per node: accumulate features into per-graph sums; lane0 counts
__global__ void __launch_bounds__(256)
k_pool(const float* __restrict__ h, const long long* __restrict__ batch,
       float* __restrict__ gsum, float* __restrict__ gcnt) {
    int n = blockIdx.x * 8 + (threadIdx.x >> 5);
    if (n >= NN) return;
    const int lane = threadIdx.x & 31;
    const int g = (int)batch[n];
    const float4 v = ((const float4*)(h + (size_t)n * HID))[lane];
    float* gs = gsum + (size_t)g * HID + lane * 4;
    atomAddF(gs + 0, v.x);
    atomAddF(gs + 1, v.y);
    atomAddF(gs + 2, v.z);
    atomAddF(gs + 3, v.w);
    if (lane == 0) atomAddF(&gcnt[g], 1.0f);
}

__global__ void k_final(const float* __restrict__ gsum, const float* __restrict__ gcnt,
                        const float* __restrict__ Wlin, const float* __restrict__ blin,
                        float* __restrict__ out) {
    int idx = blockIdx.x * blockDim.x + threadIdx.x;
    if (idx >= NG * NCLS) return;
    const int g = idx / NCLS;
    const int c = idx % NCLS;
    const float inv = 1.0f / fmaxf(gcnt[g], 1.0f);
    const float* gr = gsum + (size_t)g * HID;
    float acc = 0.f;
#pragma unroll 8
    for (int k = 0; k < HID; ++k) acc = fmaf(gr[k], Wlin[k * NCLS + c], acc);
    out[idx] = acc * inv + blin[c];
}

// ------------------------------------------------------------------ driver --
extern "C" void kernel_launch(void* const* d_in, const int* in_sizes, int n_in,
                              void* d_out, int out_size, void* d_ws, size_t ws_size,
                              hipStream_t stream) {
    const float*     x    = (const float*)d_in[0];
    const long long* ei   = (const long long*)d_in[1];   // [2, NE] int64
    const long long* bat  = (const long long*)d_in[2];
    const float*     W1   = (const float*)d_in[3];
    const float*     b1   = (const float*)d_in[4];
    const float*     W2   = (const float*)d_in[5];
    const float*     b2   = (const float*)d_in[6];
    const float*     W3   = (const float*)d_in[7];
    const float*     b3   = (const float*)d_in[8];
    const float*     Wlin = (const float*)d_in[9];
    const float*     blin = (const float*)d_in[10];
    float*           out  = (float*)d_out;

    const long long* src = ei;
    const long long* dst = ei + NE;

    float* ws    = (float*)d_ws;
    float* hbuf  = ws;                              // 12.8M
    float* agg   = ws + 12800000;                   // 12.8M
    float* deg   = ws + 25600000;                   // 100k
    float* dis   = ws + 25700000;                   // 100k
    float* selfw = ws + 25800000;                   // 100k
    float* nrm   = ws + 25900000;                   // 1.6M
    float* gsum  = ws + 27500000;                   // 640k
    float* gcnt  = ws + 28140000;                   // 5k
    float* xp    = ws + 28150000;                   // 1.6M (padded x, NN x 16)

    const int TB = 256;
    const int gNode  = (NN + TB - 1) / TB;
    const int gEdge  = (NE + TB - 1) / TB;
    const int gFeat  = (NN * HID + TB - 1) / TB;    // 50000
    const int gGemm  = (NN + 127) / 128;            // 782
    const int gScat  = (NE + 7) / 8;                // 200000 (wave per edge)
    const int gPool  = (NN + 7) / 8;                // 12500
    const int gFin   = (NG * NCLS + TB - 1) / TB;
    const int gPad   = (NN * 16 + TB - 1) / TB;

    // --- normalization coefficients (reused by all 3 layers) ---
    k_fill<<<gNode, TB, 0, stream>>>(deg, 1.0f, NN);               // self loop
    k_edge_deg<<<gEdge, TB, 0, stream>>>(dst, deg, NE);
    k_node_norm<<<gNode, TB, 0, stream>>>(deg, dis, selfw, NN);
    k_edge_coef<<<gEdge, TB, 0, stream>>>(src, dst, dis, nrm, NE);

    // --- layer 1 (K padded 11 -> 16) ---
    k_pad_x<<<gPad, TB, 0, stream>>>(x, xp);
    k_gemm_wmma<16, FIN><<<gGemm, TB, 0, stream>>>(xp, W1, hbuf, NN);
    k_combine_init<<<gFeat, TB, 0, stream>>>(hbuf, selfw, b1, agg);
    k_edge_scatter<<<gScat, TB, 0, stream>>>(hbuf, agg, src, dst, nrm, NE);
    k_relu<<<gFeat, TB, 0, stream>>>(agg);

    // --- layer 2 ---
    k_gemm_wmma<HID, HID><<<gGemm, TB, 0, stream>>>(agg, W2, hbuf, NN);
    k_combine_init<<<gFeat, TB, 0, stream>>>(hbuf, selfw, b2, agg);
    k_edge_scatter<<<gScat, TB, 0, stream>>>(hbuf, agg, src, dst, nrm, NE);
    k_relu<<<gFeat, TB, 0, stream>>>(agg);

    // --- layer 3 (no relu) ---
    k_gemm_wmma<HID, HID><<<gGemm, TB, 0, stream>>>(agg, W3, hbuf, NN);
    k_combine_init<<<gFeat, TB, 0, stream>>>(hbuf, selfw, b3, agg);
    k_edge_scatter<<<gScat, TB, 0, stream>>>(hbuf, agg, src, dst, nrm, NE);

    // --- global mean pool + linear head ---
    k_fill<<<(NG * HID + TB - 1) / TB, TB, 0, stream>>>(gsum, 0.0f, NG * HID);
    k_fill<<<(NG + TB - 1) / TB, TB, 0, stream>>>(gcnt, 0.0f, NG);
    k_pool<<<gPool, TB, 0, stream>>>(agg, bat, gsum, gcnt);
    k_final<<<gFin, TB, 0, stream>>>(gsum, gcnt, Wlin, blin, out);
}